// GATConv_000_67508295958893
// MI455X (gfx1250) — compile-verified
//
#include <hip/hip_runtime.h>
#include <math.h>

// ---------------------------------------------------------------------------
// GATConv for MI455X (gfx1250, wave32).
//   ft  = feat @ W^T                  -> WMMA f32 16x16x4 GEMM (fp32 exact path)
//   el/er = einsum(ft, attn_l/r)      -> tiny per-(node,head) kernel
//   softmax over incoming edges       -> key-encoded atomicMax + atomicAdd
//   rst = segment_sum(ft[src] * a)    -> 32-lane-per-edge gather + f32 atomics
// Edge phase is the bandwidth bottleneck (~1.6GB, L2-resident: ft=25.6MB << 192MB L2).
// ---------------------------------------------------------------------------

#define NNODES 50000
#define NEDGES 1600000
#define IN_F   256
#define HEADS  4
#define OUT_F  32
#define CO     (HEADS * OUT_F)      /* 128 output features */
#define NEG_SLOPE 0.2f

typedef __attribute__((ext_vector_type(2))) float v2f;
typedef __attribute__((ext_vector_type(8))) float v8f;

// Monotonic unsigned key for float max: preserves total order, init key 0 acts
// as -inf identity (any real float's key > 0).
__device__ __forceinline__ unsigned f2key(float x) {
  unsigned b = __float_as_uint(x);
  return (b & 0x80000000u) ? ~b : (b | 0x80000000u);
}
__device__ __forceinline__ float key2f(unsigned k) {
  unsigned b = (k & 0x80000000u) ? (k ^ 0x80000000u) : ~k;
  return __uint_as_float(b);
}
__device__ __forceinline__ float leaky(float x) {
  return x > 0.f ? x : NEG_SLOPE * x;
}

// ---------------------------------------------------------------------------
// init: zero output accumulator, reset max-keys and softmax sums
// ---------------------------------------------------------------------------
__global__ __launch_bounds__(256) void k_init(float* __restrict__ out,
                                              unsigned* __restrict__ mkey,
                                              float* __restrict__ ssum) {
  int i = blockIdx.x * 256 + threadIdx.x;
  if (i < NNODES * CO) out[i] = 0.f;
  if (i < NNODES * HEADS) { mkey[i] = 0u; ssum[i] = 0.f; }
}

// ---------------------------------------------------------------------------
// GEMM: ft[N][128] = feat[N][256] @ W^T   (W is [128][256] row-major)
// Block = 256 threads = 8 waves. Block owns 16 rows; wave w owns cols [16w,16w+16).
// A tile staged in LDS (coalesced float4 load, row stride padded to 260 floats
// -> LDS bank stride of 4, conflict-free for the 16-lane column read pattern).
// Fragment layouts per CDNA5 ISA 7.12.2:
//   A 16x4 f32 : lanes 0-15 M=lane {V0=K0,V1=K1}; lanes 16-31 M=lane-16 {K2,K3}
//   B 4x16 f32 : lanes 0-15 N=lane {V0=K0,V1=K1}; lanes 16-31 N=lane-16 {K2,K3}
//   C 16x16 f32: VGPR j = rows {j, j+8}, N = lane & 15
// ---------------------------------------------------------------------------
__global__ __launch_bounds__(256) void k_gemm(const float* __restrict__ feat,
                                              const float* __restrict__ W,
                                              float* __restrict__ ft) {
  __shared__ float As[16][IN_F + 4];          // +4 pad: bank-conflict-free cols
  const int tid    = threadIdx.x;
  const int wave   = tid >> 5;                // 0..7  (wave32)
  const int lane   = tid & 31;
  const int tile_m = blockIdx.x * 16;

  // Cooperative A-tile load: 16 rows x 256 floats = 1024 float4, 4 per thread.
  for (int i = tid; i < 16 * (IN_F / 4); i += 256) {
    const int r  = i / (IN_F / 4);
    const int c4 = i % (IN_F / 4);
    float4 v = make_float4(0.f, 0.f, 0.f, 0.f);
    const int row = tile_m + r;
    if (row < NNODES) v = ((const float4*)(feat + (size_t)row * IN_F))[c4];
    ((float4*)&As[r][0])[c4] = v;
  }
  __syncthreads();

  const int koff = (lane < 16) ? 0 : 2;       // K sub-offset for this lane half
  const int arow = lane & 15;                 // A row / B col within tile
  const int col  = wave * 16 + arow;          // global output column (<128)
  const float* wrow = W + (size_t)col * IN_F; // W[col][*] == (W^T)[*][col]

  v8f c = {};
  for (int k0 = 0; k0 < IN_F; k0 += 4) {
    v2f a, b;
    a.x = As[arow][k0 + koff];
    a.y = As[arow][k0 + koff + 1];
    b.x = wrow[k0 + koff];
    b.y = wrow[k0 + koff + 1];
    c = __builtin_amdgcn_wmma_f32_16x16x4_f32(
        /*neg_a=*/false, a, /*neg_b=*/false, b,
        /*c_mod=*/(short)0, c, /*reuse_a=*/false, /*reuse_b=*/false);
  }

  // Store C tile.
  const int mbase  = (lane < 16) ? 0 : 8;
  const int outcol = col;
#pragma unroll
  for (int j = 0; j < 8; ++j) {
    const int row = tile_m + mbase + j;
    if (row < NNODES) ft[(size_t)row * CO + outcol] = c[j];
  }
}

// ---------------------------------------------------------------------------
// el[n,h] = dot(ft[n,h,:], attn[0,h,:32]) ; er with attn[0,h,32:64]
// ---------------------------------------------------------------------------
__global__ __launch_bounds__(256) void k_eler(const float* __restrict__ ft,
                                              const float* __restrict__ attn,
                                              float* __restrict__ el,
                                              float* __restrict__ er) {
  int i = blockIdx.x * 256 + threadIdx.x;     // over N*HEADS
  if (i >= NNODES * HEADS) return;
  const int n = i / HEADS, h = i % HEADS;
  const float* f  = ft + (size_t)n * CO + h * OUT_F;
  const float* al = attn + h * 2 * OUT_F;
  const float* ar = al + OUT_F;
  float sl = 0.f, sr = 0.f;
#pragma unroll
  for (int d = 0; d < OUT_F; ++d) {
    const float v = f[d];
    sl += v * al[d];
    sr += v * ar[d];
  }
  el[i] = sl;
  er[i] = sr;
}

// ---------------------------------------------------------------------------
// segment_max(e, dst) via unsigned-key atomicMax
// ---------------------------------------------------------------------------
__global__ __launch_bounds__(256) void k_edge_max(const int* __restrict__ src,
                                                  const int* __restrict__ dst,
                                                  const float* __restrict__ el,
                                                  const float* __restrict__ er,
                                                  unsigned* __restrict__ mkey) {
  int i = blockIdx.x * 256 + threadIdx.x;     // over E*HEADS
  if (i >= NEDGES * HEADS) return;
  const int e = i >> 2, h = i & 3;
  const int s = src[e], d = dst[e];
  const float x = leaky(el[s * HEADS + h] + er[d * HEADS + h]);
  atomicMax(&mkey[d * HEADS + h], f2key(x));
}

// ---------------------------------------------------------------------------
// s[dst,h] += exp(e - m[dst,h])   (e recomputed: cheaper than 25.6MB spill)
// ---------------------------------------------------------------------------
__global__ __launch_bounds__(256) void k_edge_sum(const int* __restrict__ src,
                                                  const int* __restrict__ dst,
                                                  const float* __restrict__ el,
                                                  const float* __restrict__ er,
                                                  const unsigned* __restrict__ mkey,
                                                  float* __restrict__ ssum) {
  int i = blockIdx.x * 256 + threadIdx.x;     // over E*HEADS
  if (i >= NEDGES * HEADS) return;
  const int e = i >> 2, h = i & 3;
  const int s = src[e], d = dst[e];
  const float x = leaky(el[s * HEADS + h] + er[d * HEADS + h]);
  const float m = key2f(mkey[d * HEADS + h]);
  atomicAdd(&ssum[d * HEADS + h], expf(x - m));
}

// ---------------------------------------------------------------------------
// rst[dst,h,:] += ft[src,h,:] * a     (one 32-lane group per edge; lane = feat d,
// loop over 4 heads -> contiguous 128-float row gather + 128 f32 atomics)
// ---------------------------------------------------------------------------
__global__ __launch_bounds__(256) void k_edge_agg(const int* __restrict__ src,
                                                  const int* __restrict__ dst,
                                                  const float* __restrict__ el,
                                                  const float* __restrict__ er,
                                                  const unsigned* __restrict__ mkey,
                                                  const float* __restrict__ ssum,
                                                  const float* __restrict__ ft,
                                                  float* __restrict__ out) {
  const int lane = threadIdx.x & 31;
  const int eg   = blockIdx.x * 8 + (threadIdx.x >> 5);   // 8 waves/block, wave32
  if (eg >= NEDGES) return;
  const int s = src[eg], d = dst[eg];

  const float* fsrc = ft + (size_t)s * CO;
  __builtin_prefetch(fsrc, 0, 0);             // global_prefetch_b8 on source row

#pragma unroll
  for (int h = 0; h < HEADS; ++h) {
    const float x = leaky(el[s * HEADS + h] + er[d * HEADS + h]);
    const float m = key2f(mkey[d * HEADS + h]);
    const float a = expf(x - m) / ssum[d * HEADS + h];
    const float v = fsrc[h * OUT_F + lane] * a;
    atomicAdd(&out[(size_t)d * CO + h * OUT_F + lane], v);
  }
}

// ---------------------------------------------------------------------------
// launcher
// ---------------------------------------------------------------------------
extern "C" void kernel_launch(void* const* d_in, const int* in_sizes, int n_in,
                              void* d_out, int out_size, void* d_ws, size_t ws_size,
                              hipStream_t stream) {
  (void)in_sizes; (void)n_in; (void)out_size; (void)ws_size;

  const float* feat = (const float*)d_in[0];
  const int*   src  = (const int*)d_in[1];
  const int*   dst  = (const int*)d_in[2];
  const float* W    = (const float*)d_in[3];
  const float* attn = (const float*)d_in[4];
  float* out = (float*)d_out;

  // Workspace layout (~28.8 MB total)
  char* ws = (char*)d_ws;
  float*    ft   = (float*)ws;    ws += (size_t)NNODES * CO * sizeof(float);     // 25.6 MB
  float*    el   = (float*)ws;    ws += (size_t)NNODES * HEADS * sizeof(float);  // 0.8 MB
  float*    er   = (float*)ws;    ws += (size_t)NNODES * HEADS * sizeof(float);
  unsigned* mkey = (unsigned*)ws; ws += (size_t)NNODES * HEADS * sizeof(unsigned);
  float*    ssum = (float*)ws;    ws += (size_t)NNODES * HEADS * sizeof(float);

  k_init<<<(NNODES * CO + 255) / 256, 256, 0, stream>>>(out, mkey, ssum);
  k_gemm<<<NNODES / 16, 256, 0, stream>>>(feat, W, ft);                 // 3125 blocks
  k_eler<<<(NNODES * HEADS + 255) / 256, 256, 0, stream>>>(ft, attn, el, er);
  k_edge_max<<<(NEDGES * HEADS + 255) / 256, 256, 0, stream>>>(src, dst, el, er, mkey);
  k_edge_sum<<<(NEDGES * HEADS + 255) / 256, 256, 0, stream>>>(src, dst, el, er, mkey, ssum);
  k_edge_agg<<<NEDGES / 8, 256, 0, stream>>>(src, dst, el, er, mkey, ssum, ft, out);
}